// SpearmanLoss_34299608826515
// MI455X (gfx1250) — compile-verified
//
#include <hip/hip_runtime.h>
#include <hip/hip_bf16.h>
#include <math.h>

typedef __attribute__((ext_vector_type(2))) float v2f;
typedef __attribute__((ext_vector_type(8))) float v8f;

#define SR_REG 1.0f

// ---------------------------------------------------------------------------
// Kernel 1: soft_rank for one input per block.
//   - bitonic sort (descending) of (theta, orig_index) in LDS
//   - PAV isotonic regression (non-increasing fit) of y_i = s_i - (n - i),
//     stack-based, double-precision block sums (single thread; n=4096 serial
//     dependency chain, negligible vs launch latency)
//   - scatter ranks_sorted[i] = s_i - v_i back to original positions
// Dynamic LDS layout (bytes, n = 4096 -> 96 KB total, fine in 320 KB/WGP):
//   [0      , 4n ) float  s[n]
//   [4n     , 8n ) int    perm[n]
//   [8n     , 16n) double bsum[n]
//   [16n    , 20n) int    bcnt[n]
//   [20n    , 24n) int    bstart[n]
// ---------------------------------------------------------------------------
__global__ __launch_bounds__(1024) void soft_rank_kernel(
    const float* __restrict__ xa, const float* __restrict__ xb,
    float* __restrict__ oa, float* __restrict__ ob, int n) {
  const float* __restrict__ x   = (blockIdx.x == 0) ? xa : xb;
  float* __restrict__       out = (blockIdx.x == 0) ? oa : ob;

  extern __shared__ char smem[];
  float*  s      = (float*)(smem);
  int*    perm   = (int*)(smem + (size_t)4 * n);
  double* bsum   = (double*)(smem + (size_t)8 * n);
  int*    bcnt   = (int*)(smem + (size_t)16 * n);
  int*    bstart = (int*)(smem + (size_t)20 * n);

  const int tid = threadIdx.x;
  const int nt  = blockDim.x;

  // Load theta = x / reg, init permutation.
  for (int i = tid; i < n; i += nt) {
    s[i]    = x[i] / SR_REG;
    perm[i] = i;
  }
  __syncthreads();

  // Bitonic sort, descending (final pass k==n has (i&k)==0 for all i).
  for (int k = 2; k <= n; k <<= 1) {
    for (int j = k >> 1; j > 0; j >>= 1) {
      for (int i = tid; i < n; i += nt) {
        int ixj = i ^ j;
        if (ixj > i) {
          bool desc = ((i & k) == 0);
          float va = s[i], vb = s[ixj];
          bool do_swap = desc ? (va < vb) : (va > vb);
          if (do_swap) {
            s[i] = vb; s[ixj] = va;
            int pa = perm[i]; perm[i] = perm[ixj]; perm[ixj] = pa;
          }
        }
      }
      __syncthreads();
    }
  }

  // PAV: project y onto { v_1 >= v_2 >= ... >= v_n }, merge when the new
  // (right) block's mean exceeds the previous (left) block's mean.
  if (tid == 0) {
    int top = -1;
    for (int i = 0; i < n; ++i) {
      double csum  = (double)s[i] - (double)(n - i); // y_i = s_i - w_i, w_i = n-i
      int    ccnt  = 1;
      int    cbeg  = i;
      while (top >= 0 && csum * (double)bcnt[top] >= bsum[top] * (double)ccnt) {
        csum += bsum[top];
        ccnt += bcnt[top];
        cbeg  = bstart[top];
        --top;
      }
      ++top;
      bsum[top]   = csum;
      bcnt[top]   = ccnt;
      bstart[top] = cbeg;
    }
    // Expand block means and scatter ranks to original positions.
    for (int j = 0; j <= top; ++j) {
      float vm  = (float)(bsum[j] / (double)bcnt[j]);
      int   beg = bstart[j];
      int   end = beg + bcnt[j];
      for (int i = beg; i < end; ++i) {
        out[perm[i]] = s[i] - vm;   // ranks_sorted scattered through perm
      }
    }
  }
}

// ---------------------------------------------------------------------------
// Kernel 2: centered Gram via V_WMMA_F32_16X16X4_F32, one wave32.
//   Means computed first in double (wave shuffle reduction) so the WMMA
//   accumulates centered values (~O(1) magnitude) -- no catastrophic
//   cancellation in f32.
//   A tile (16x4 f32): lane m in 0..15 holds row m, K = {0,1}; lanes 16..31
//   hold row m-16, K = {2,3}. B (4x16) mirrors with N<->M, so for the
//   symmetric operand X the per-lane register contents of A and B coincide.
//   Row 0 = centered p, row 1 = centered t, rows 2..15 = 0.
//   C layout: lane (M<16? N : N) ... G[M][N]: lanes 0..15 carry M=vgpr,
//   lanes 16..31 carry M=vgpr+8; we dump C to LDS and read G[0][0], G[0][1],
//   G[1][1].
// ---------------------------------------------------------------------------
__global__ void spearman_wmma_kernel(const float* __restrict__ p,
                                     const float* __restrict__ t,
                                     float* __restrict__ out, int n) {
  const int lane = threadIdx.x; // 0..31, one wave

  // Phase 1: means in double.
  double sp = 0.0, st = 0.0;
  for (int i = lane; i < n; i += 32) {
    sp += (double)p[i];
    st += (double)t[i];
  }
  for (int off = 16; off > 0; off >>= 1) {
    sp += __shfl_xor(sp, off, 32);
    st += __shfl_xor(st, off, 32);
  }
  const float mp = (float)(sp / (double)n);
  const float mt = (float)(st / (double)n);

  // Phase 2: Gram of centered rows via f32 WMMA, K stepped by 4.
  const int r    = lane & 15;              // matrix row this lane feeds
  const int koff = (lane < 16) ? 0 : 2;    // K sub-slice per lane half
  v8f acc = {};
  for (int k = 0; k < n; k += 4) {
    float2 vp = *(const float2*)(p + k + koff);
    float2 vt = *(const float2*)(t + k + koff);
    if (k + 128 < n) {
      __builtin_prefetch(p + k + 128, 0, 3);
      __builtin_prefetch(t + k + 128, 0, 3);
    }
    v2f a;
    a.x = (r == 0) ? (vp.x - mp) : (r == 1) ? (vt.x - mt) : 0.0f;
    a.y = (r == 0) ? (vp.y - mp) : (r == 1) ? (vt.y - mt) : 0.0f;
    // D = A * A^T + C  (symmetric operand: same lane registers for A and B)
    acc = __builtin_amdgcn_wmma_f32_16x16x4_f32(
        /*neg_a=*/false, a, /*neg_b=*/false, a,
        /*c_mod=*/(short)0, acc, /*reuse_a=*/false, /*reuse_b=*/false);
  }

  __shared__ float G[16][16];
#pragma unroll
  for (int v = 0; v < 8; ++v) {
    int row = (lane < 16) ? v : (v + 8);
    G[row][lane & 15] = acc[v];
  }
  __syncthreads();

  if (lane == 0) {
    float spp = G[0][0]; // sum p'^2
    float spt = G[0][1]; // sum p'*t'
    float stt = G[1][1]; // sum t'^2
    out[0] = 1.0f - spt / (sqrtf(spp) * sqrtf(stt));
  }
}

// ---------------------------------------------------------------------------
extern "C" void kernel_launch(void* const* d_in, const int* in_sizes, int n_in,
                              void* d_out, int out_size, void* d_ws, size_t ws_size,
                              hipStream_t stream) {
  const float* pred = (const float*)d_in[0];
  const float* targ = (const float*)d_in[1];
  const int n = in_sizes[0]; // 4096 (power of two; bitonic network assumes it)

  float* wsP = (float*)d_ws;       // soft ranks of pred
  float* wsT = wsP + n;            // soft ranks of target

  const size_t shmem = (size_t)24 * (size_t)n; // 96 KB for n=4096

  soft_rank_kernel<<<2, 1024, shmem, stream>>>(pred, targ, wsP, wsT, n);
  spearman_wmma_kernel<<<1, 32, 0, stream>>>(wsP, wsT, (float*)d_out, n);
}